// WindowAttention_34222299414649
// MI455X (gfx1250) — compile-verified
//
#include <hip/hip_runtime.h>
#include <hip/hip_bf16.h>

// Swin window attention, fully fused: one workgroup per 7x7 window.
// fp32 end-to-end using V_WMMA_F32_16X16X4_F32 (16x16 tiles, K-step 4).
//
//   B=64, H=W=56, C=128, HEADS=4, WS=7, N=49 (padded to 64), HD=32,
//   nW = 8*8 = 64 windows/image, grid = B*nW = 4096 blocks, 256 thr (8 waves).
//
// Round-2 changes vs round-1:
//  * each wave owns one M-tile and a PAIR of N-tiles -> A fragment loaded once
//    per two WMMAs, and two independent accumulation chains (better XDL issue).
//  * double-buffered weight staging (lds_w0/lds_w1) with register prefetch so
//    L2->LDS latency overlaps the WMMA phase (we are 1 workgroup/WGP at 218KB LDS).
//  * attn@v uses a K-split accumulator pair to break the serial D->C chain.

typedef float v2f __attribute__((ext_vector_type(2)));
typedef float v8f __attribute__((ext_vector_type(8)));

#define WMMA_F32X4(a, b, c) \
    __builtin_amdgcn_wmma_f32_16x16x4_f32(false, (a), false, (b), (short)0, (c), false, false)

// LDS strides padded so 16-lane row-strided fragment reads hit distinct banks:
// bank = (stride*row + k) % 64 with stride % 64 == 4.
#define LDX 132   // x tile / attn-out tile: 64 x 128 (+4 pad)
#define LDW 132   // transposed weight chunk: 64(n) x 128(k) (+4 pad)
#define LDQ 388   // qkv: 64 x 384 (+4 pad)
#define LDS_SC 68 // scores: 64 x 64 (+4 pad)

#define SMEM_FLOATS (64 * LDX + 2 * 64 * LDW + 64 * LDQ + 64 * LDS_SC)

__global__ __launch_bounds__(256) void swin_window_attn_kernel(
    const float* __restrict__ x,        // [64,56,56,128]
    const float* __restrict__ qkv_w,    // [128,384]
    const float* __restrict__ qkv_b,    // [384]
    const float* __restrict__ table,    // [169,4]
    const float* __restrict__ proj_w,   // [128,128]
    const float* __restrict__ proj_b,   // [128]
    const float* __restrict__ mask,     // [64,1,4,49,49]
    const int*   __restrict__ rel_idx,  // [2401]
    float*       __restrict__ out)      // [64,56,56,128]
{
    extern __shared__ float smem[];
    float* lds_x  = smem;                    // 64 x LDX   (x tile, later attn-out)
    float* lds_w0 = lds_x + 64 * LDX;        // 64 x LDW   (weight chunk, buf 0)
    float* lds_w1 = lds_w0 + 64 * LDW;       // 64 x LDW   (weight chunk, buf 1)
    float* lds_q  = lds_w1 + 64 * LDW;       // 64 x LDQ   (qkv: q|k|v per token)
    float* lds_s  = lds_q + 64 * LDQ;        // 64 x LDS_SC (scores per head)
    float* wbuf[2] = { lds_w0, lds_w1 };

    const int tid  = threadIdx.x;
    const int lane = tid & 31;
    const int wave = tid >> 5;
    const int half = lane >> 4;   // selects K pair within fragment
    const int l16  = lane & 15;   // row-within-tile (A) / col-within-tile (B,D)

    const int bid = blockIdx.x;        // 0..4095
    const int b   = bid >> 6;          // image
    const int wi  = bid & 63;          // window within image
    const int h0  = (wi >> 3) * 7;     // window origin row
    const int w0  = (wi & 7) * 7;      // window origin col

    // per-wave tile assignment for 4x4-tile stages: one M-tile, two N-tiles
    const int w_mt  = wave >> 1;
    const int w_nt0 = (wave & 1) * 2;

    // ---- stage 0: load x window (49 x 128) into lds_x, zero-pad rows 49..63
    for (int i = tid; i < 64 * 32; i += 256) {
        const int row = i >> 5;
        const int c4  = (i & 31) * 4;
        float4 v = make_float4(0.f, 0.f, 0.f, 0.f);
        if (row < 49) {
            const int hh = h0 + row / 7;
            const int ww = w0 + row % 7;
            v = *(const float4*)(x + (((size_t)b * 56 + hh) * 56 + ww) * 128 + c4);
        }
        float* dst = lds_x + row * LDX + c4;
        dst[0] = v.x; dst[1] = v.y; dst[2] = v.z; dst[3] = v.w;
    }
    // stage qkv_w chunk 0 transposed -> wbuf[0][n][k]   (concurrent with x load)
    for (int i = tid; i < 128 * 16; i += 256) {
        const int n4 = (i & 15) * 4, k = i >> 4;
        float4 v = *(const float4*)(qkv_w + k * 384 + n4);
        float* d = wbuf[0] + n4 * LDW + k;
        d[0] = v.x; d[LDW] = v.y; d[2 * LDW] = v.z; d[3 * LDW] = v.w;
    }
    __syncthreads();

    // ---- stage 1: qkv = x @ qkv_w + qkv_b  (M=64pad, K=128, N=384 in 6 chunks)
    const float qscale = 0.17677669529663687f;  // 32^-0.5
    for (int nc = 0; nc < 6; ++nc) {
        const float* cw = wbuf[nc & 1];
        // register-prefetch next weight chunk (overlaps WMMA phase)
        float4 pre[8];
        if (nc < 5) {
            for (int j = 0; j < 8; ++j) {
                const int i = tid + j * 256;
                const int n4 = (i & 15) * 4, k = i >> 4;
                pre[j] = *(const float4*)(qkv_w + k * 384 + (nc + 1) * 64 + n4);
            }
        }
        v8f acc0 = {}, acc1 = {};
        const float* arow = lds_x + (w_mt * 16 + l16) * LDX + 2 * half;
        const float* br0  = cw + (w_nt0 * 16 + l16) * LDW + 2 * half;
        const float* br1  = br0 + 16 * LDW;
        for (int k0 = 0; k0 < 128; k0 += 4) {
            v2f a  = *(const v2f*)(arow + k0);
            v2f b0 = *(const v2f*)(br0 + k0);
            v2f b1 = *(const v2f*)(br1 + k0);
            acc0 = WMMA_F32X4(a, b0, acc0);
            acc1 = WMMA_F32X4(a, b1, acc1);
        }
        {
            const int c0 = nc * 64 + w_nt0 * 16 + l16;   // 0..383
            const int c1 = c0 + 16;
            const float bs0 = qkv_b[c0], bs1 = qkv_b[c1];
            const float s0 = (c0 < 128) ? qscale : 1.0f; // q gets * hd^-0.5
            const float s1 = (c1 < 128) ? qscale : 1.0f;
            for (int r = 0; r < 8; ++r) {
                const int m = w_mt * 16 + r + 8 * half;
                lds_q[m * LDQ + c0] = (acc0[r] + bs0) * s0;
                lds_q[m * LDQ + c1] = (acc1[r] + bs1) * s1;
            }
        }
        if (nc < 5) {
            float* nw = wbuf[(nc + 1) & 1];
            for (int j = 0; j < 8; ++j) {
                const int i = tid + j * 256;
                const int n4 = (i & 15) * 4, k = i >> 4;
                float* d = nw + n4 * LDW + k;
                d[0] = pre[j].x; d[LDW] = pre[j].y;
                d[2 * LDW] = pre[j].z; d[3 * LDW] = pre[j].w;
            }
        }
        __syncthreads();
    }

    // stage proj_w chunk 0 into wbuf[0] now; it completes while attention runs
    for (int i = tid; i < 128 * 16; i += 256) {
        const int n4 = (i & 15) * 4, k = i >> 4;
        float4 v = *(const float4*)(proj_w + k * 128 + n4);
        float* d = wbuf[0] + n4 * LDW + k;
        d[0] = v.x; d[LDW] = v.y; d[2 * LDW] = v.z; d[3 * LDW] = v.w;
    }

    // ---- stage 2: attention per head
    for (int h = 0; h < 4; ++h) {
        __syncthreads();  // qkv ready (h==0) / lds_s reuse across heads
        // scores = q @ k^T  (K = 32), two N-tiles per wave sharing A
        {
            v8f acc0 = {}, acc1 = {};
            const float* arow = lds_q + (w_mt * 16 + l16) * LDQ + h * 32 + 2 * half;        // q
            const float* br0  = lds_q + (w_nt0 * 16 + l16) * LDQ + 128 + h * 32 + 2 * half; // k
            const float* br1  = br0 + 16 * LDQ;
            for (int k0 = 0; k0 < 32; k0 += 4) {
                v2f a  = *(const v2f*)(arow + k0);
                v2f b0 = *(const v2f*)(br0 + k0);
                v2f b1 = *(const v2f*)(br1 + k0);
                acc0 = WMMA_F32X4(a, b0, acc0);
                acc1 = WMMA_F32X4(a, b1, acc1);
            }
            const int n0 = w_nt0 * 16 + l16;
            const int n1 = n0 + 16;
            for (int r = 0; r < 8; ++r) {
                const int m = w_mt * 16 + r + 8 * half;
                float v0 = -1e30f, v1 = -1e30f;  // padded rows/cols die pre-softmax
                if (m < 49 && n0 < 49)
                    v0 = acc0[r] + table[rel_idx[m * 49 + n0] * 4 + h]
                       + mask[(((size_t)wi * 4 + h) * 49 + m) * 49 + n0];
                if (m < 49 && n1 < 49)
                    v1 = acc1[r] + table[rel_idx[m * 49 + n1] * 4 + h]
                       + mask[(((size_t)wi * 4 + h) * 49 + m) * 49 + n1];
                lds_s[m * LDS_SC + n0] = v0;
                lds_s[m * LDS_SC + n1] = v1;
            }
        }
        __syncthreads();
        // row softmax (padded cols -> exactly 0)
        if (tid < 64) {
            float* row = lds_s + tid * LDS_SC;
            float mx = row[0];
            for (int n = 1; n < 64; ++n) mx = fmaxf(mx, row[n]);
            float sum = 0.f;
            for (int n = 0; n < 64; ++n) { float e = __expf(row[n] - mx); row[n] = e; sum += e; }
            const float inv = 1.f / sum;
            for (int n = 0; n < 64; ++n) row[n] *= inv;
        }
        __syncthreads();
        // out_h = attn @ v  (M=64pad, K=64pad, N=32) -> lds_x[:, h*32 : h*32+32]
        // K split into two accumulator chains (k<32 / k>=32), summed at the end.
        {
            const int mt = wave >> 1, nt = wave & 1;
            v8f accA = {}, accB = {};
            const float* arow = lds_s + (mt * 16 + l16) * LDS_SC + 2 * half;
            const float* vcol = lds_q + 256 + h * 32 + nt * 16 + l16;
            for (int k0 = 0; k0 < 32; k0 += 4) {
                const int kkA = k0 + 2 * half;
                const int kkB = kkA + 32;
                v2f a0 = *(const v2f*)(arow + k0);
                v2f a1 = *(const v2f*)(arow + k0 + 32);
                v2f b0, b1;
                b0.x = vcol[(size_t)kkA * LDQ];
                b0.y = vcol[(size_t)(kkA + 1) * LDQ];
                b1.x = vcol[(size_t)kkB * LDQ];
                b1.y = vcol[(size_t)(kkB + 1) * LDQ];
                accA = WMMA_F32X4(a0, b0, accA);
                accB = WMMA_F32X4(a1, b1, accB);
            }
            v8f acc = accA + accB;
            const int col = h * 32 + nt * 16 + l16;
            for (int r = 0; r < 8; ++r) {
                const int m = mt * 16 + r + 8 * half;
                lds_x[m * LDX + col] = acc[r];
            }
        }
    }
    __syncthreads();  // attn-out complete; proj_w chunk 0 staged in wbuf[0]

    // ---- stage 3: out = attn_out @ proj_w + proj_b, window-reverse scatter
    for (int nc = 0; nc < 2; ++nc) {
        const float* cw = wbuf[nc];
        float4 pre[8];
        if (nc == 0) {  // prefetch proj_w chunk 1
            for (int j = 0; j < 8; ++j) {
                const int i = tid + j * 256;
                const int n4 = (i & 15) * 4, k = i >> 4;
                pre[j] = *(const float4*)(proj_w + k * 128 + 64 + n4);
            }
        }
        v8f acc0 = {}, acc1 = {};
        const float* arow = lds_x + (w_mt * 16 + l16) * LDX + 2 * half;
        const float* br0  = cw + (w_nt0 * 16 + l16) * LDW + 2 * half;
        const float* br1  = br0 + 16 * LDW;
        for (int k0 = 0; k0 < 128; k0 += 4) {
            v2f a  = *(const v2f*)(arow + k0);
            v2f b0 = *(const v2f*)(br0 + k0);
            v2f b1 = *(const v2f*)(br1 + k0);
            acc0 = WMMA_F32X4(a, b0, acc0);
            acc1 = WMMA_F32X4(a, b1, acc1);
        }
        {
            const int c0 = nc * 64 + w_nt0 * 16 + l16;
            const int c1 = c0 + 16;
            const float p0 = proj_b[c0], p1 = proj_b[c1];
            for (int r = 0; r < 8; ++r) {
                const int m = w_mt * 16 + r + 8 * half;
                if (m < 49) {
                    const int hh = h0 + m / 7;
                    const int ww = w0 + m % 7;
                    float* o = out + (((size_t)b * 56 + hh) * 56 + ww) * 128;
                    o[c0] = acc0[r] + p0;
                    o[c1] = acc1[r] + p1;
                }
            }
        }
        if (nc == 0) {
            float* nw = wbuf[1];
            for (int j = 0; j < 8; ++j) {
                const int i = tid + j * 256;
                const int n4 = (i & 15) * 4, k = i >> 4;
                float* d = nw + n4 * LDW + k;
                d[0] = pre[j].x; d[LDW] = pre[j].y;
                d[2 * LDW] = pre[j].z; d[3 * LDW] = pre[j].w;
            }
            __syncthreads();
        }
    }
}

extern "C" void kernel_launch(void* const* d_in, const int* in_sizes, int n_in,
                              void* d_out, int out_size, void* d_ws, size_t ws_size,
                              hipStream_t stream) {
    const float* x      = (const float*)d_in[0];
    const float* qkv_w  = (const float*)d_in[1];
    const float* qkv_b  = (const float*)d_in[2];
    const float* table  = (const float*)d_in[3];
    const float* proj_w = (const float*)d_in[4];
    const float* proj_b = (const float*)d_in[5];
    const float* mask   = (const float*)d_in[6];
    const int*   relidx = (const int*)d_in[7];
    // d_in[8]/d_in[9] = height/width (compile-time constants 56)
    float* out = (float*)d_out;

    const int grid = 64 * 64;                        // B * nW
    const size_t smem = SMEM_FLOATS * sizeof(float); // ~213 KB (fits 320 KB WGP LDS)
    swin_window_attn_kernel<<<grid, 256, smem, stream>>>(
        x, qkv_w, qkv_b, table, proj_w, proj_b, mask, relidx, out);
}